// MLA_7524782702766
// MI455X (gfx1250) — compile-verified
//
#include <hip/hip_runtime.h>
#include <hip/hip_bf16.h>
#include <math.h>

// ---------------- problem constants (from reference) ----------------
#define HID    5120
#define QLORA  1536
#define NH     128
#define NOPE   128
#define ROPE   64
#define KVLORA 512
#define QHD    192            // NOPE + ROPE
#define VHD    128
#define BB     2              // batch
#define QLEN   512
#define PAST   512
#define KTOT   1024           // PAST + QLEN
#define EPSF   1e-6f

typedef float v2f __attribute__((ext_vector_type(2)));
typedef float v8f __attribute__((ext_vector_type(8)));

// gfx1250 async global->LDS copy (ASYNCcnt-tracked, no VGPR data round trip)
__device__ __forceinline__ void async_g2l_b128(void* lds_ptr, const float* gptr)
{
    unsigned lds_addr = (unsigned)(uintptr_t)lds_ptr;   // low 32b = LDS offset
    asm volatile("global_load_async_to_lds_b128 %0, %1, off"
                 :: "v"(lds_addr), "v"(gptr) : "memory");
}
__device__ __forceinline__ void wait_async0()
{
    asm volatile("s_wait_asynccnt 0x0" ::: "memory");
}

// =====================================================================
// Tiled fp32 WMMA GEMM:  C = A (MxK, row major) * B (NxK, row major)^T
// Block tile 64(M) x 256(N), 8 waves, each wave 32x64 via 2x4 WMMA frags.
// Register-pipelined staging: next K-chunk's global loads are in flight
// while the current chunk's WMMAs run.
// MODE 0: plain row-major C (with N guard)
// MODE 1: q epilogue  -> (row=b*512+qi, col=h*192+d) to (b,h,qi,d)
// MODE 2: kv epilogue -> col=h*256+j ; j<128 -> key cache, j>=128 -> value
// =====================================================================
#define BM 64
#define BN 256
#define KC 32

template <int MODE>
__global__ __launch_bounds__(256) void gemm_wmma(
    const float* __restrict__ A, const float* __restrict__ Bw,
    float* __restrict__ C0, float* __restrict__ C1,
    int M, int Nn, int K)
{
    __shared__ float As[BM][KC + 1];
    __shared__ float Bs[BN][KC + 1];

    const int tid   = threadIdx.x;
    const int lane  = tid & 31;
    const int wave  = tid >> 5;
    const int wm    = wave & 1;    // 2 waves along M
    const int wn    = wave >> 1;   // 4 waves along N (64 cols each)
    const int m0    = blockIdx.y * BM;
    const int n0    = blockIdx.x * BN;
    const int lmod  = lane & 15;
    const int lhalf = lane >> 4;
    const int klo   = lhalf * 2;   // A/B fragment k-offset per ISA layout

    v8f acc[2][4];
    #pragma unroll
    for (int i = 0; i < 2; ++i)
        #pragma unroll
        for (int j = 0; j < 4; ++j)
            #pragma unroll
            for (int e = 0; e < 8; ++e) acc[i][j][e] = 0.0f;

    // staging registers: A 64x32 = 512 f4 (2/thr), B 256x32 = 2048 f4 (8/thr)
    float4 regA[2], regB[8];
    const int ar[2]  = { (tid + 0) >> 3, (tid + 256) >> 3 };
    const int ac4    = (tid & 7) << 2;

    auto load_regs = [&](int k0) {
        #pragma unroll
        for (int it = 0; it < 2; ++it)
            regA[it] = *reinterpret_cast<const float4*>(
                &A[(size_t)(m0 + ar[it]) * K + k0 + ac4]);
        #pragma unroll
        for (int it = 0; it < 8; ++it) {
            int f  = tid + it * 256;
            int r  = f >> 3;
            int gr = n0 + r; if (gr >= Nn) gr = Nn - 1;   // N-tail clamp
            regB[it] = *reinterpret_cast<const float4*>(
                &Bw[(size_t)gr * K + k0 + ((f & 7) << 2)]);
        }
    };
    auto store_lds = [&]() {
        #pragma unroll
        for (int it = 0; it < 2; ++it) {
            As[ar[it]][ac4 + 0] = regA[it].x; As[ar[it]][ac4 + 1] = regA[it].y;
            As[ar[it]][ac4 + 2] = regA[it].z; As[ar[it]][ac4 + 3] = regA[it].w;
        }
        #pragma unroll
        for (int it = 0; it < 8; ++it) {
            int f = tid + it * 256;
            int r = f >> 3, c4 = (f & 7) << 2;
            Bs[r][c4 + 0] = regB[it].x; Bs[r][c4 + 1] = regB[it].y;
            Bs[r][c4 + 2] = regB[it].z; Bs[r][c4 + 3] = regB[it].w;
        }
    };

    load_regs(0);
    for (int k0 = 0; k0 < K; k0 += KC) {
        store_lds();
        __syncthreads();
        if (k0 + KC < K) load_regs(k0 + KC);   // overlap with WMMA below
        if (k0 + 2 * KC < K)                   // warm L2 one chunk further out
            __builtin_prefetch(&A[(size_t)(m0 + (tid >> 3)) * K + k0 + 2 * KC], 0, 0);

        #pragma unroll
        for (int kk = 0; kk < KC; kk += 4) {
            v2f af[2], bf[4];
            #pragma unroll
            for (int mi = 0; mi < 2; ++mi) {
                int m = wm * 32 + mi * 16 + lmod;
                af[mi].x = As[m][kk + klo];
                af[mi].y = As[m][kk + klo + 1];
            }
            #pragma unroll
            for (int ni = 0; ni < 4; ++ni) {
                int n = wn * 64 + ni * 16 + lmod;
                bf[ni].x = Bs[n][kk + klo];
                bf[ni].y = Bs[n][kk + klo + 1];
            }
            #pragma unroll
            for (int mi = 0; mi < 2; ++mi)
                #pragma unroll
                for (int ni = 0; ni < 4; ++ni)
                    acc[mi][ni] = __builtin_amdgcn_wmma_f32_16x16x4_f32(
                        false, af[mi], false, bf[ni], (short)0,
                        acc[mi][ni], false, false);
        }
        __syncthreads();
    }

    // ---- epilogue: C frag layout -> lanes 0-15 rows 0..7, 16-31 rows 8..15
    const int mbase = lhalf * 8;
    #pragma unroll
    for (int mi = 0; mi < 2; ++mi) {
        #pragma unroll
        for (int ni = 0; ni < 4; ++ni) {
            #pragma unroll
            for (int r = 0; r < 8; ++r) {
                const int m = m0 + wm * 32 + mi * 16 + mbase + r;
                const int n = n0 + wn * 64 + ni * 16 + lmod;
                const float val = acc[mi][ni][r];
                if (MODE == 0) {
                    if (n < Nn) C0[(size_t)m * Nn + n] = val;
                } else if (MODE == 1) {
                    // q: (b*512+qi, h*192+d) -> (b,h,qi,d)
                    const int b = m >> 9, qi = m & 511;
                    const int h = n / QHD, d = n % QHD;
                    C0[(((size_t)(b * NH + h)) * QLEN + qi) * QHD + d] = val;
                } else {
                    // kv: (b*512+qi, h*256+j) -> key/value cache (new half)
                    const int b = m >> 9, qi = m & 511;
                    const int h = n >> 8, j = n & 255;
                    if (j < NOPE)
                        C0[(((size_t)(b * NH + h)) * KTOT + PAST + qi) * QHD + j] = val;
                    else
                        C1[(((size_t)(b * NH + h)) * KTOT + PAST + qi) * VHD + (j - NOPE)] = val;
                }
            }
        }
    }
}

// =====================================================================
// RMSNorm (one row per block):  y = g * x * rsqrt(mean(x^2) + eps)
// =====================================================================
__global__ __launch_bounds__(256) void rmsnorm_kernel(
    const float* __restrict__ in, int in_stride,
    float* __restrict__ out, int out_stride,
    const float* __restrict__ g, int cols)
{
    __shared__ float red[8];
    const int row = blockIdx.x;
    const float* x = in + (size_t)row * in_stride;
    float* y       = out + (size_t)row * out_stride;

    float ss = 0.0f;
    for (int i = threadIdx.x; i < cols; i += 256) { float v = x[i]; ss += v * v; }
    #pragma unroll
    for (int off = 16; off > 0; off >>= 1) ss += __shfl_xor(ss, off, 32);
    if ((threadIdx.x & 31) == 0) red[threadIdx.x >> 5] = ss;
    __syncthreads();
    if (threadIdx.x == 0) {
        float t = 0.0f;
        #pragma unroll
        for (int i = 0; i < 8; ++i) t += red[i];
        red[0] = t;
    }
    __syncthreads();
    const float rs = rsqrtf(red[0] / (float)cols + EPSF);
    for (int i = threadIdx.x; i < cols; i += 256) y[i] = g[i] * (x[i] * rs);
}

// =====================================================================
// KV-cache assembly: copy past K/V into the concat output, broadcast k_pe
// =====================================================================
__global__ __launch_bounds__(256) void copy_past_key(
    const float* __restrict__ pk, float* __restrict__ ko)
{
    size_t idx = (size_t)blockIdx.x * 256 + threadIdx.x;          // float4 idx
    const size_t total = (size_t)BB * NH * PAST * QHD / 4;
    if (idx >= total) return;
    const float4 v = reinterpret_cast<const float4*>(pk)[idx];
    const size_t e  = idx * 4;
    const size_t d  = e % QHD;
    const size_t k  = (e / QHD) % PAST;
    const size_t bh = e / ((size_t)QHD * PAST);
    *reinterpret_cast<float4*>(&ko[(bh * KTOT + k) * QHD + d]) = v;
}

__global__ __launch_bounds__(256) void copy_past_value(
    const float* __restrict__ pv, float* __restrict__ vo)
{
    size_t idx = (size_t)blockIdx.x * 256 + threadIdx.x;
    const size_t total = (size_t)BB * NH * PAST * VHD / 4;
    if (idx >= total) return;
    const float4 v = reinterpret_cast<const float4*>(pv)[idx];
    const size_t e  = idx * 4;
    const size_t d  = e % VHD;
    const size_t k  = (e / VHD) % PAST;
    const size_t bh = e / ((size_t)VHD * PAST);
    *reinterpret_cast<float4*>(&vo[(bh * KTOT + k) * VHD + d]) = v;
}

__global__ __launch_bounds__(256) void bcast_kpe(
    const float* __restrict__ ckv, float* __restrict__ ko)
{
    size_t idx = (size_t)blockIdx.x * 256 + threadIdx.x;          // float4 idx
    const size_t total = (size_t)BB * NH * QLEN * (ROPE / 4);
    if (idx >= total) return;
    const size_t r4 = idx % (ROPE / 4);
    const size_t qi = (idx / (ROPE / 4)) % QLEN;
    const size_t h  = (idx / ((ROPE / 4) * QLEN)) % NH;
    const size_t b  = idx / ((size_t)(ROPE / 4) * QLEN * NH);
    const float4 v  = *reinterpret_cast<const float4*>(
        &ckv[(b * QLEN + qi) * (KVLORA + ROPE) + KVLORA + r4 * 4]);
    *reinterpret_cast<float4*>(
        &ko[((b * NH + h) * KTOT + PAST + qi) * QHD + NOPE + r4 * 4]) = v;
}

// =====================================================================
// Flash attention: block = 8 waves = 128 q rows of one (b,h).
// K/V tiles double-buffered in LDS and filled with ASYNCcnt-tracked
// global_load_async_to_lds_b128 while the previous tile's WMMAs run.
// =====================================================================
__global__ __launch_bounds__(256) void mla_flash_attn(
    const float* __restrict__ Q,    // (b,h,512,192)
    const float* __restrict__ Kst,  // (b,h,1024,192)
    const float* __restrict__ Vst,  // (b,h,1024,128)
    const float* __restrict__ mask, // (b,1,512,1024)
    float* __restrict__ O,          // (b,h,512,128)
    float scale)
{
    __shared__ float Ks[2][16][QHD + 4];   // 2 x 16 x 196
    __shared__ float Vs[2][16][VHD + 4];   // 2 x 16 x 132
    __shared__ float Ps[8][16][17];        // per-wave P redistribution

    const int tid   = threadIdx.x;
    const int lane  = tid & 31;
    const int w     = tid >> 5;
    const int lmod  = lane & 15;
    const int lhalf = lane >> 4;
    const int klo   = lhalf * 2;
    const int mbase = lhalf * 8;

    const int blk = blockIdx.x;
    const int qb  = blk & 3;                 // 4 q-blocks of 128 rows
    const int h   = (blk >> 2) & (NH - 1);
    const int b   = blk >> 9;
    const int q0w = qb * 128 + w * 16;       // this wave's first q row

    const float* kbase = Kst + ((size_t)(b * NH + h)) * KTOT * QHD;
    const float* vbase = Vst + ((size_t)(b * NH + h)) * KTOT * VHD;
    const float* mptr  = mask + (size_t)b * QLEN * KTOT;

    // ---- load Q A-fragments (48 d-chunks of 4) into registers ----
    const float* qptr = Q + (((size_t)(b * NH + h)) * QLEN + q0w + lmod) * QHD;
    v2f aq[QHD / 4];
    #pragma unroll
    for (int dc = 0; dc < QHD / 4; ++dc) {
        const float2 v = *reinterpret_cast<const float2*>(&qptr[dc * 4 + klo]);
        aq[dc].x = v.x; aq[dc].y = v.y;
    }

    v8f o[VHD / 16];
    float mrow[8], lrow[8];
    #pragma unroll
    for (int ni = 0; ni < VHD / 16; ++ni)
        #pragma unroll
        for (int e = 0; e < 8; ++e) o[ni][e] = 0.0f;
    #pragma unroll
    for (int r = 0; r < 8; ++r) { mrow[r] = -__builtin_inff(); lrow[r] = 0.0f; }

    // async stage of one 16-key K/V tile into LDS buffer `sel`
    auto stage_tile = [&](int kt, int sel) {
        const int k0 = kt * 16;
        #pragma unroll
        for (int it = 0; it < 3; ++it) {           // K: 768 float4
            int f = tid + it * 256;
            int r = f / 48, c4 = (f % 48) * 4;
            async_g2l_b128(&Ks[sel][r][c4], &kbase[(size_t)(k0 + r) * QHD + c4]);
        }
        #pragma unroll
        for (int it = 0; it < 2; ++it) {           // V: 512 float4
            int f = tid + it * 256;
            int r = f >> 5, c4 = (f & 31) * 4;
            async_g2l_b128(&Vs[sel][r][c4], &vbase[(size_t)(k0 + r) * VHD + c4]);
        }
    };

    stage_tile(0, 0);
    wait_async0();
    __syncthreads();

    const int NT = KTOT / 16;
    for (int kt = 0; kt < NT; ++kt) {
        const int cur = kt & 1;
        const int k0  = kt * 16;
        if (kt + 1 < NT) stage_tile(kt + 1, cur ^ 1);  // overlap with compute

        // ---- scores S = Q * K^T (16q x 16k) ----
        v8f s;
        #pragma unroll
        for (int e = 0; e < 8; ++e) s[e] = 0.0f;
        #pragma unroll
        for (int dc = 0; dc < QHD / 4; ++dc) {
            v2f bk;
            bk.x = Ks[cur][lmod][dc * 4 + klo];
            bk.y = Ks[cur][lmod][dc * 4 + klo + 1];
            s = __builtin_amdgcn_wmma_f32_16x16x4_f32(
                false, aq[dc], false, bk, (short)0, s, false, false);
        }

        // ---- online softmax (per C-frag row, 16-lane butterflies) ----
        #pragma unroll
        for (int r = 0; r < 8; ++r) {
            const int qrow = q0w + mbase + r;
            float sv = s[r] * scale + mptr[(size_t)qrow * KTOT + k0 + lmod];
            float rowmax = sv;
            #pragma unroll
            for (int off = 1; off < 16; off <<= 1)
                rowmax = fmaxf(rowmax, __shfl_xor(rowmax, off, 32));
            const float mnew  = fmaxf(mrow[r], rowmax);
            const float alpha = __expf(mrow[r] - mnew);
            const float pv    = __expf(sv - mnew);
            float rowsum = pv;
            #pragma unroll
            for (int off = 1; off < 16; off <<= 1)
                rowsum += __shfl_xor(rowsum, off, 32);
            lrow[r] = lrow[r] * alpha + rowsum;
            mrow[r] = mnew;
            #pragma unroll
            for (int ni = 0; ni < VHD / 16; ++ni) o[ni][r] *= alpha;
            Ps[w][mbase + r][lmod] = pv;   // C-layout -> LDS (per-wave buffer)
        }

        // ---- O += P * V  (contraction over 16 keys, 4 chunks of 4) ----
        #pragma unroll
        for (int kc = 0; kc < 4; ++kc) {
            v2f ap;
            ap.x = Ps[w][lmod][kc * 4 + klo];
            ap.y = Ps[w][lmod][kc * 4 + klo + 1];
            #pragma unroll
            for (int ni = 0; ni < VHD / 16; ++ni) {
                v2f bv;
                bv.x = Vs[cur][kc * 4 + klo][ni * 16 + lmod];
                bv.y = Vs[cur][kc * 4 + klo + 1][ni * 16 + lmod];
                o[ni] = __builtin_amdgcn_wmma_f32_16x16x4_f32(
                    false, ap, false, bv, (short)0, o[ni], false, false);
            }
        }

        wait_async0();       // next tile landed in LDS (our wave's loads)
        __syncthreads();     // ...and everyone else's
    }

    // ---- finalize and store ----
    #pragma unroll
    for (int r = 0; r < 8; ++r) {
        const float inv = 1.0f / lrow[r];
        const int qrow = q0w + mbase + r;
        float* optr = O + (((size_t)(b * NH + h)) * QLEN + qrow) * VHD;
        #pragma unroll
        for (int ni = 0; ni < VHD / 16; ++ni)
            optr[ni * 16 + lmod] = o[ni][r] * inv;
    }
}

// =====================================================================
// launcher
// =====================================================================
extern "C" void kernel_launch(void* const* d_in, const int* in_sizes, int n_in,
                              void* d_out, int out_size, void* d_ws, size_t ws_size,
                              hipStream_t stream)
{
    const float* hs     = (const float*)d_in[0];  // (2,512,5120)
    const float* mask   = (const float*)d_in[1];  // (2,1,512,1024)
    const float* pk     = (const float*)d_in[2];  // (2,128,512,192)
    const float* pv     = (const float*)d_in[3];  // (2,128,512,128)
    const float* w_q_a  = (const float*)d_in[4];  // (1536,5120)
    const float* g_q_a  = (const float*)d_in[5];  // (1536)
    const float* w_q_b  = (const float*)d_in[6];  // (24576,1536)
    const float* w_kv_a = (const float*)d_in[7];  // (576,5120)
    const float* g_kv_a = (const float*)d_in[8];  // (512)
    const float* w_kv_b = (const float*)d_in[9];  // (32768,512)

    float* attn_out = (float*)d_out;                               // (2,128,512,128)
    float* key_out  = attn_out + (size_t)BB * NH * QLEN * VHD;     // (2,128,1024,192)
    float* val_out  = key_out  + (size_t)BB * NH * KTOT * QHD;     // (2,128,1024,128)

    // workspace: qa (6MB) | ckv (2.25MB) | ckvn (2MB) | q (96MB)  ~= 111 MB
    float* qa   = (float*)d_ws;
    float* ckv  = qa   + (size_t)(BB * QLEN) * QLORA;
    float* ckvn = ckv  + (size_t)(BB * QLEN) * (KVLORA + ROPE);
    float* q_ws = ckvn + (size_t)(BB * QLEN) * KVLORA;

    const int Mrows = BB * QLEN;   // 1024
    dim3 blk(256);

    // 1) qa = hs @ w_q_a^T           (1024 x 5120 x 1536)
    gemm_wmma<0><<<dim3(QLORA / BN, Mrows / BM), blk, 0, stream>>>(
        hs, w_q_a, qa, nullptr, Mrows, QLORA, HID);
    // 2) ckv = hs @ w_kv_a^T         (1024 x 5120 x 576)
    gemm_wmma<0><<<dim3((KVLORA + ROPE + BN - 1) / BN, Mrows / BM), blk, 0, stream>>>(
        hs, w_kv_a, ckv, nullptr, Mrows, KVLORA + ROPE, HID);
    // 3) rmsnorm (qa in place; ckv[:, :512] -> ckvn)
    rmsnorm_kernel<<<Mrows, blk, 0, stream>>>(qa, QLORA, qa, QLORA, g_q_a, QLORA);
    rmsnorm_kernel<<<Mrows, blk, 0, stream>>>(ckv, KVLORA + ROPE, ckvn, KVLORA, g_kv_a, KVLORA);
    // 4) q = qa @ w_q_b^T -> (b,h,q,192) layout directly
    gemm_wmma<1><<<dim3(NH * QHD / BN, Mrows / BM), blk, 0, stream>>>(
        qa, w_q_b, q_ws, nullptr, Mrows, NH * QHD, QLORA);
    // 5) kv = ckvn @ w_kv_b^T -> scattered straight into K/V cache outputs
    gemm_wmma<2><<<dim3(NH * (NOPE + VHD) / BN, Mrows / BM), blk, 0, stream>>>(
        ckvn, w_kv_b, key_out, val_out, Mrows, NH * (NOPE + VHD), KVLORA);
    // 6) cache assembly: past K/V copies + k_pe broadcast over heads
    {
        size_t n4 = (size_t)BB * NH * PAST * QHD / 4;
        copy_past_key<<<(unsigned)((n4 + 255) / 256), blk, 0, stream>>>(pk, key_out);
        n4 = (size_t)BB * NH * PAST * VHD / 4;
        copy_past_value<<<(unsigned)((n4 + 255) / 256), blk, 0, stream>>>(pv, val_out);
        n4 = (size_t)BB * NH * QLEN * (ROPE / 4);
        bcast_kpe<<<(unsigned)((n4 + 255) / 256), blk, 0, stream>>>(ckv, key_out);
    }
    // 7) flash attention over the assembled cache
    const float scale = 0.707f / sqrtf((float)QHD);
    mla_flash_attn<<<BB * NH * (QLEN / 128), blk, 0, stream>>>(
        q_ws, key_out, val_out, mask, attn_out, scale);
}